// ApproachNet_regression_view_fps_23682449670881
// MI455X (gfx1250) — compile-verified
//
#include <hip/hip_runtime.h>

typedef float v2f __attribute__((ext_vector_type(2)));
typedef float v8f __attribute__((ext_vector_type(8)));
typedef unsigned int v4u __attribute__((ext_vector_type(4)));
typedef int v8i __attribute__((ext_vector_type(8)));
typedef int v4i __attribute__((ext_vector_type(4)));

#define NUM_VIEW   300
#define NUM_SAMPLE 1024
#define BN_EPS     1e-5f
#define GRASP_THRESH 0.1f
#define BATCH 4
#define NPTS  20000
#define CH    256
#define NTILES32 (NPTS / 32)     // 625 head tiles per batch
#define STILES (NUM_SAMPLE / 16) // 64

// ---- flat output offsets (float32, reference tuple order) ----
#define OFF_OBJ   0
#define OFF_GRASP (OFF_OBJ   + BATCH * 2 * NPTS)          // 160000
#define OFF_GXYZ  (OFF_GRASP + BATCH * NPTS)              // 240000
#define OFF_GINDS (OFF_GXYZ  + BATCH * NUM_SAMPLE * 3)    // 252288
#define OFF_GFEAT (OFF_GINDS + BATCH * NUM_SAMPLE)        // 256384
#define OFF_FP2   (OFF_GFEAT + BATCH * CH * NUM_SAMPLE)   // 1304960
#define OFF_VPXYZ (OFF_FP2   + BATCH * NUM_SAMPLE)        // 1309056
#define OFF_TOPV  (OFF_VPXYZ + BATCH * NUM_SAMPLE * 3)    // 1321344
#define OFF_ROT   (OFF_TOPV  + BATCH * NUM_SAMPLE)        // 1325440

// =====================================================================
// Kernel 1: fused  h = relu(BN(W1·X + b1));  graspable = W2·h + b2
// 32-column N-tile per workgroup; X tile staged by the Tensor Data Mover
// (tensor_load_to_lds, TENSORcnt). 16 waves; wave w -> rows 16w..16w+15,
// two accumulators (cols 0-15 / 16-31) reusing each A-fragment.
// fp32-exact WMMA (v_wmma_f32_16x16x4_f32).
// =====================================================================
__global__ __launch_bounds__(512)
void head_kernel(const float* __restrict__ feat,
                 const float* __restrict__ w1, const float* __restrict__ b1,
                 const float* __restrict__ g,  const float* __restrict__ bb,
                 const float* __restrict__ bm, const float* __restrict__ bv,
                 const float* __restrict__ w2, const float* __restrict__ b2,
                 float* __restrict__ out, int* __restrict__ mask)
{
    __shared__ float Xs[CH][32];     // 32 KB feature tile (TDM destination)
    __shared__ float sc[CH], sh[CH]; // folded BN scale / shift
    __shared__ float red[3][32];     // 3x256 second-layer reduction

    const int tile = blockIdx.x;
    const int b    = tile / NTILES32;
    const int n0   = (tile % NTILES32) * 32;
    const int tid  = threadIdx.x;
    const int wave = tid >> 5;
    const int lane = tid & 31;

    const float* fb = feat + (size_t)b * CH * NPTS;

    // ---- TDM: DMA the strided 2D tile [CH rows x 32 floats, row stride
    //      NPTS] from global into LDS. Issued once by wave 0. ----
    if (wave == 0) {
        unsigned long long gaddr = (unsigned long long)(const void*)(fb + n0);
        unsigned int ldsoff = (unsigned int)(unsigned long long)(const void*)&Xs[0][0];
        v4u g0;
        g0[0] = 1u;                                   // count=1 valid descriptor
        g0[1] = ldsoff;                               // lds_addr (bytes)
        g0[2] = (unsigned int)gaddr;                  // global_addr[31:0]
        g0[3] = (unsigned int)((gaddr >> 32) & 0x01FFFFFFu) | (2u << 30); // [56:32] | type=2
        v8i g1;
        g1[0] = (int)(2u << 16);                      // data_size = 2 (4 bytes)
        g1[1] = (int)(((unsigned)NPTS & 0xFFFFu) << 16);        // tensor_dim0[15:0]
        g1[2] = (int)(((unsigned)NPTS >> 16) | (((unsigned)CH & 0xFFFFu) << 16)); // dim0 hi | dim1 lo
        g1[3] = (int)(((unsigned)CH >> 16) | (32u << 16));      // dim1 hi | tile_dim0=32
        g1[4] = (int)CH;                              // tile_dim1=256, tile_dim2=0
        g1[5] = (int)NPTS;                            // tensor_dim0_stride[31:0]
        g1[6] = 0;                                    // stride0 hi | stride1 lo
        g1[7] = 0;
        v4i g2 = {0, 0, 0, 0};                        // 2-D tile: dims 2+ unused
        v4i g3 = {0, 0, 0, 0};
        v8i g4 = {0, 0, 0, 0, 0, 0, 0, 0};            // extra group (6-arg toolchain form)
        __builtin_amdgcn_tensor_load_to_lds(g0, g1, g2, g3, g4, 0);
        __builtin_amdgcn_s_wait_tensorcnt(0);
    }
    if (tid < CH) {
        float s = g[tid] * rsqrtf(bv[tid] + BN_EPS);
        sc[tid] = s;
        sh[tid] = (b1[tid] - bm[tid]) * s + bb[tid];
    }
    if (tid < 96) red[tid / 32][tid & 31] = b2[tid / 32];
    __syncthreads();

    // ---- WMMA K-loop: two 16x16 tiles per wave, shared A-fragment ----
    const int m0    = wave * 16;
    const int arow  = m0 + (lane & 15);
    const int khalf = (lane >> 4) << 1;   // lanes 0-15 -> K+0/1, lanes 16-31 -> K+2/3
    const int ncol  = lane & 15;
    v8f acc0 = {};
    v8f acc1 = {};
    for (int k = 0; k < CH; k += 4) {
        v2f a, bf0, bf1;
        a.x   = w1[(size_t)arow * CH + k + khalf];
        a.y   = w1[(size_t)arow * CH + k + khalf + 1];
        bf0.x = Xs[k + khalf][ncol];
        bf0.y = Xs[k + khalf + 1][ncol];
        bf1.x = Xs[k + khalf][ncol + 16];
        bf1.y = Xs[k + khalf + 1][ncol + 16];
        acc0 = __builtin_amdgcn_wmma_f32_16x16x4_f32(false, a, false, bf0,
                                                     (short)0, acc0, false, false);
        acc1 = __builtin_amdgcn_wmma_f32_16x16x4_f32(false, a, false, bf1,
                                                     (short)0, acc1, false, false);
    }

    // ---- BN + ReLU + partial 3x256 projection (h never hits memory) ----
    const int rofs = m0 + ((lane >> 4) << 3); // C/D layout: VGPR r -> row rofs+r
    float p0 = 0.f, p1 = 0.f, p2 = 0.f;
    float q0 = 0.f, q1 = 0.f, q2 = 0.f;
#pragma unroll
    for (int r = 0; r < 8; ++r) {
        int row = rofs + r;
        float s = sc[row], t = sh[row];
        float w20 = w2[0 * CH + row], w21 = w2[1 * CH + row], w22 = w2[2 * CH + row];
        float h0 = acc0[r] * s + t; h0 = h0 > 0.f ? h0 : 0.f;
        float h1 = acc1[r] * s + t; h1 = h1 > 0.f ? h1 : 0.f;
        p0 += w20 * h0; p1 += w21 * h0; p2 += w22 * h0;
        q0 += w20 * h1; q1 += w21 * h1; q2 += w22 * h1;
    }
    atomicAdd(&red[0][ncol], p0);
    atomicAdd(&red[1][ncol], p1);
    atomicAdd(&red[2][ncol], p2);
    atomicAdd(&red[0][ncol + 16], q0);
    atomicAdd(&red[1][ncol + 16], q1);
    atomicAdd(&red[2][ncol + 16], q2);
    __syncthreads();

    if (tid < 32) {
        int n = n0 + tid;
        float o0 = red[0][tid], o1 = red[1][tid], gr = red[2][tid];
        out[OFF_OBJ + ((size_t)b * 2 + 0) * NPTS + n] = o0;
        out[OFF_OBJ + ((size_t)b * 2 + 1) * NPTS + n] = o1;
        out[OFF_GRASP + (size_t)b * NPTS + n]         = gr;
        mask[b * NPTS + n] = (o1 > o0 && gr > GRASP_THRESH) ? 1 : 0;
    }
}

// =====================================================================
// Kernel 2: masked farthest-point sampling (serial bottleneck).
// One persistent 1024-thread workgroup per batch; dist[] lives in LDS
// (80 KB of the 320 KB WGP pool). Masked-out points encoded as dist<0.
// =====================================================================
__global__ __launch_bounds__(1024)
void fps_kernel(const float* __restrict__ xyz, const int* __restrict__ mask,
                float* __restrict__ out, int* __restrict__ inds_ws)
{
    __shared__ float dist[NPTS];   // 80 KB
    __shared__ float redv[1024];
    __shared__ int   redi[1024];
    __shared__ float pxyz[3];

    const int b   = blockIdx.x;
    const int tid = threadIdx.x;
    const float* xb = xyz + (size_t)b * NPTS * 3;
    const int*   mb = mask + b * NPTS;

    // init dist, find first masked index (jnp.argmax of bool mask)
    int firstLocal = NPTS;
    for (int i = tid; i < NPTS; i += 1024) {
        int mk = mb[i];
        dist[i] = mk ? 1e10f : -1e10f;
        if (mk && i < firstLocal) firstLocal = i;
    }
    redi[tid] = firstLocal;
    __syncthreads();
    for (int s = 512; s > 0; s >>= 1) {
        if (tid < s) redi[tid] = min(redi[tid], redi[tid + s]);
        __syncthreads();
    }
    int last = redi[0];
    if (last >= NPTS) last = 0;
    __syncthreads();

    for (int s = 0; s < NUM_SAMPLE; ++s) {
        if (tid == 0) {
            inds_ws[b * NUM_SAMPLE + s]            = last;
            out[OFF_GINDS + b * NUM_SAMPLE + s]    = (float)last;
            float px = xb[last * 3 + 0], py = xb[last * 3 + 1], pz = xb[last * 3 + 2];
            pxyz[0] = px; pxyz[1] = py; pxyz[2] = pz;
            size_t o = OFF_GXYZ + ((size_t)b * NUM_SAMPLE + s) * 3;
            out[o + 0] = px; out[o + 1] = py; out[o + 2] = pz;
            out[OFF_FP2 + b * NUM_SAMPLE + s] = out[OFF_GRASP + (size_t)b * NPTS + last];
        }
        __syncthreads();
        float px = pxyz[0], py = pxyz[1], pz = pxyz[2];

        float bestv = -3e38f; int besti = 0;
        for (int i = tid; i < NPTS; i += 1024) {
            float dcur = dist[i];
            if (dcur >= 0.f) {            // masked-in point
                float dx = xb[i * 3 + 0] - px;
                float dy = xb[i * 3 + 1] - py;
                float dz = xb[i * 3 + 2] - pz;
                float d  = dx * dx + dy * dy + dz * dz;
                dcur = fminf(dcur, d);
                dist[i] = dcur;
            }
            if (dcur > bestv || (dcur == bestv && i < besti)) { bestv = dcur; besti = i; }
        }
        redv[tid] = bestv; redi[tid] = besti;
        __syncthreads();
        for (int r = 512; r > 0; r >>= 1) {
            if (tid < r) {
                float ov = redv[tid + r]; int oi = redi[tid + r];
                if (ov > redv[tid] || (ov == redv[tid] && oi < redi[tid])) {
                    redv[tid] = ov; redi[tid] = oi;
                }
            }
            __syncthreads();
        }
        last = redi[0];
        __syncthreads();
    }
}

// =====================================================================
// Kernel 3: gather sampled features (emitting graspable_features),
// fused conv1x1+BN+ReLU (WMMA) + 3x256 projection -> vp_xyz,
// 300-view argmax + rotation matrices. Gather stays VGPR-indexed
// (TDM gather mode indexes rows; this gather is strided columns).
// =====================================================================
__global__ __launch_bounds__(512)
void approach_kernel(const float* __restrict__ feat,
                     const int* __restrict__ inds_ws,
                     const float* __restrict__ w1, const float* __restrict__ b1,
                     const float* __restrict__ g,  const float* __restrict__ bb,
                     const float* __restrict__ bm, const float* __restrict__ bv,
                     const float* __restrict__ w2, const float* __restrict__ b2,
                     float* __restrict__ out)
{
    __shared__ float Xs[CH][16];
    __shared__ float sc[CH], sh[CH];
    __shared__ float red[3][16];
    __shared__ float views[NUM_VIEW][3];
    __shared__ int   sidx[16];

    const int tile = blockIdx.x;
    const int b    = tile / STILES;
    const int s0   = (tile % STILES) * 16;
    const int tid  = threadIdx.x;
    const int wave = tid >> 5;
    const int lane = tid & 31;

    if (tid < 16) sidx[tid] = inds_ws[b * NUM_SAMPLE + s0 + tid];
    if (tid < CH) {
        float s = g[tid] * rsqrtf(bv[tid] + BN_EPS);
        sc[tid] = s;
        sh[tid] = (b1[tid] - bm[tid]) * s + bb[tid];
    }
    if (tid < 48) red[tid / 16][tid & 15] = b2[tid / 16];
    if (tid < NUM_VIEW) {
        // match numpy's float64 Fibonacci-sphere view generation
        double phi = 0.61803398874989484820;
        double z   = (2.0 * tid + 1.0) / (double)NUM_VIEW - 1.0;
        double r   = sqrt(fmax(1.0 - z * z, 0.0));
        double ang = 2.0 * 3.14159265358979323846 * (double)tid * phi;
        views[tid][0] = (float)(r * cos(ang));
        views[tid][1] = (float)(r * sin(ang));
        views[tid][2] = (float)z;
    }
    __syncthreads();

    const float* fb = feat + (size_t)b * CH * NPTS;
    for (int i = tid; i < CH * 16; i += 512) {
        int c = i >> 4, n = i & 15;
        float val = fb[(size_t)c * NPTS + sidx[n]];
        Xs[c][n] = val;
        out[OFF_GFEAT + ((size_t)b * CH + c) * NUM_SAMPLE + s0 + n] = val;
    }
    __syncthreads();

    const int m0    = wave * 16;
    const int arow  = m0 + (lane & 15);
    const int khalf = (lane >> 4) << 1;
    const int ncol  = lane & 15;
    v8f acc = {};
    for (int k = 0; k < CH; k += 4) {
        v2f a, bfr;
        a.x   = w1[(size_t)arow * CH + k + khalf];
        a.y   = w1[(size_t)arow * CH + k + khalf + 1];
        bfr.x = Xs[k + khalf][ncol];
        bfr.y = Xs[k + khalf + 1][ncol];
        acc = __builtin_amdgcn_wmma_f32_16x16x4_f32(false, a, false, bfr,
                                                    (short)0, acc, false, false);
    }
    const int rofs = m0 + ((lane >> 4) << 3);
    float p0 = 0.f, p1 = 0.f, p2 = 0.f;
#pragma unroll
    for (int r = 0; r < 8; ++r) {
        int row = rofs + r;
        float h = acc[r] * sc[row] + sh[row];
        h = h > 0.f ? h : 0.f;
        p0 += w2[0 * CH + row] * h;
        p1 += w2[1 * CH + row] * h;
        p2 += w2[2 * CH + row] * h;
    }
    atomicAdd(&red[0][ncol], p0);
    atomicAdd(&red[1][ncol], p1);
    atomicAdd(&red[2][ncol], p2);
    __syncthreads();

    if (tid < 16) {
        int sIdx = s0 + tid;
        float vx = red[0][tid], vy = red[1][tid], vz = red[2][tid];
        size_t o = OFF_VPXYZ + ((size_t)b * NUM_SAMPLE + sIdx) * 3;
        out[o + 0] = vx; out[o + 1] = vy; out[o + 2] = vz;

        float npn = sqrtf(vx * vx + vy * vy + vz * vz);
        float best = -3e38f; int bi = 0;
        for (int vv = 0; vv < NUM_VIEW; ++vv) {
            float wx = views[vv][0], wy = views[vv][1], wz = views[vv][2];
            float nv = sqrtf(wx * wx + wy * wy + wz * wz);
            float sv = (wx * vx + wy * vy + wz * vz) / fmaxf(nv * npn, 1e-8f);
            if (sv > best) { best = sv; bi = vv; }   // first-index tie-break
        }
        out[OFF_TOPV + b * NUM_SAMPLE + sIdx] = (float)bi;

        // viewpoint_to_matrix(-vp)
        float ax0 = -vx, ax1 = -vy, ax2 = -vz;
        float ay0 = -ax1, ay1 = ax0, ay2 = 0.f;
        float ny = sqrtf(ay0 * ay0 + ay1 * ay1 + ay2 * ay2);
        if (ny == 0.f) { ay0 = 0.f; ay1 = 1.f; ay2 = 0.f; }
        float nx = sqrtf(ax0 * ax0 + ax1 * ax1 + ax2 * ax2);
        ax0 /= nx; ax1 /= nx; ax2 /= nx;
        float ny2 = sqrtf(ay0 * ay0 + ay1 * ay1 + ay2 * ay2);
        ay0 /= ny2; ay1 /= ny2; ay2 /= ny2;
        float az0 = ax1 * ay2 - ax2 * ay1;
        float az1 = ax2 * ay0 - ax0 * ay2;
        float az2 = ax0 * ay1 - ax1 * ay0;
        size_t ro = OFF_ROT + ((size_t)b * NUM_SAMPLE + sIdx) * 9;
        out[ro + 0] = ax0; out[ro + 1] = ay0; out[ro + 2] = az0;
        out[ro + 3] = ax1; out[ro + 4] = ay1; out[ro + 5] = az1;
        out[ro + 6] = ax2; out[ro + 7] = ay2; out[ro + 8] = az2;
    }
}

extern "C" void kernel_launch(void* const* d_in, const int* in_sizes, int n_in,
                              void* d_out, int out_size, void* d_ws, size_t ws_size,
                              hipStream_t stream)
{
    const float* seed_xyz      = (const float*)d_in[0];
    const float* seed_features = (const float*)d_in[1];
    const float* gh_w1   = (const float*)d_in[2];
    const float* gh_b1   = (const float*)d_in[3];
    const float* gh_bn_g = (const float*)d_in[4];
    const float* gh_bn_b = (const float*)d_in[5];
    const float* gh_bn_m = (const float*)d_in[6];
    const float* gh_bn_v = (const float*)d_in[7];
    const float* gh_w2   = (const float*)d_in[8];
    const float* gh_b2   = (const float*)d_in[9];
    const float* c1_w    = (const float*)d_in[10];
    const float* c1_b    = (const float*)d_in[11];
    const float* bn1_g   = (const float*)d_in[12];
    const float* bn1_b   = (const float*)d_in[13];
    const float* bn1_m   = (const float*)d_in[14];
    const float* bn1_v   = (const float*)d_in[15];
    const float* c2_w    = (const float*)d_in[16];
    const float* c2_b    = (const float*)d_in[17];

    float* out = (float*)d_out;
    int* mask  = (int*)d_ws;                 // B*N ints
    int* inds  = mask + BATCH * NPTS;        // B*1024 ints

    head_kernel<<<BATCH * NTILES32, 512, 0, stream>>>(
        seed_features, gh_w1, gh_b1, gh_bn_g, gh_bn_b, gh_bn_m, gh_bn_v,
        gh_w2, gh_b2, out, mask);

    fps_kernel<<<BATCH, 1024, 0, stream>>>(seed_xyz, mask, out, inds);

    approach_kernel<<<BATCH * STILES, 512, 0, stream>>>(
        seed_features, inds, c1_w, c1_b, bn1_g, bn1_b, bn1_m, bn1_v,
        c2_w, c2_b, out);
}